// PAN_Event_4_52982716563758
// MI455X (gfx1250) — compile-verified
//
#include <hip/hip_runtime.h>
#include <hip/hip_bf16.h>

typedef __attribute__((ext_vector_type(2))) float v2f;
typedef __attribute__((ext_vector_type(8))) float v8f;

// Problem constants (from setup_inputs)
#define Bn   2
#define Cc   320
#define Hh   128
#define Ww   256
#define XC   64
#define MAXD 48
#define NG   40
#define CPG  8
#define MC   (4*XC)   // 256 merged channels

// ---------------------------------------------------------------------------
// Kernel 1: groupwise-correlation cost volume, LDS-tiled.
// Block handles (b, h, 64-wide w tile); channels processed in 4 chunks of 80
// (10 groups each). Ref staged with float4 (16B-aligned); volume written with
// non-temporal stores (write-once stream, keep it out of L2).
// ---------------------------------------------------------------------------
__global__ __launch_bounds__(256) void gwc_volume_kernel(
    const float* __restrict__ ref, const float* __restrict__ tgt,
    float* __restrict__ vol)
{
    __shared__ float srefs[80 * 64];    // [ch_local][w_local]
    __shared__ float stgts[80 * 112];   // [ch_local][w_local - 47 .. +64]

    const int blk = blockIdx.x;
    const int wt  = blk & 3;                 // 4 w-tiles of 64
    const int h   = (blk >> 2) & (Hh - 1);
    const int b   = blk >> 9;                // /(4*128)
    const int w0  = wt * 64;

    const int tid = threadIdx.x;
    const int w   = tid & 63;                // lane's w within tile
    const int pd  = tid >> 6;                // 0..3

    for (int gc = 0; gc < 4; ++gc) {
        const int c0 = gc * 80;
        __syncthreads();   // protect previous chunk's reads
        // --- stage ref tile: 80 rows x 16 float4 (16B-aligned) ---
        {
            float4* sdst = (float4*)srefs;
            for (int i = tid; i < 80 * 16; i += 256) {
                const int cl = i >> 4, q = i & 15;
                const float* src = ref +
                    (((long)(b * Cc + c0 + cl)) * Hh + h) * Ww + w0 + q * 4;
                sdst[i] = *(const float4*)src;
            }
        }
        // --- stage tgt tile (covers w0-47 .. w0+63), unaligned -> scalar ---
        for (int i = tid; i < 80 * 112; i += 256) {
            const int cl = i / 112, wl = i % 112;
            const int gw = w0 - 47 + wl;
            float v = 0.f;
            if (wl < 111 && (unsigned)gw < (unsigned)Ww)
                v = tgt[(((long)(b * Cc + c0 + cl)) * Hh + h) * Ww + gw];
            stgts[i] = v;
        }
        __syncthreads();
        // --- all (group, disparity) pairs for this chunk ---
        for (int p = pd; p < 10 * MAXD; p += 4) {
            const int gl = p / MAXD;         // group within chunk
            const int d  = p % MAXD;
            const int rb = (gl * CPG) * 64 + w;
            const int tb = (gl * CPG) * 112 + (w + 47 - d);
            float acc = 0.f;
#pragma unroll
            for (int cc = 0; cc < CPG; ++cc)
                acc += srefs[rb + cc * 64] * stgts[tb + cc * 112];
            const int gw = w0 + w;
            const float outv = (gw >= d) ? acc * (1.0f / CPG) : 0.f;
            const int g = gc * 10 + gl;
            __builtin_nontemporal_store(outv,
                &vol[((((long)(b * NG + g)) * MAXD + d) * Hh + h) * Ww + gw]);
        }
    }
}

// ---------------------------------------------------------------------------
// Kernel 2: DDPM dynamic conv (dil 1/3/5, shared per-pixel 3x3 kernel) and
// assembly of merged = concat(x, br1, br3, br5) into workspace.
// ---------------------------------------------------------------------------
__global__ __launch_bounds__(256) void dynconv_kernel(
    const float* __restrict__ x, const float* __restrict__ ker,
    float* __restrict__ merged)
{
    const int idx = blockIdx.x * 256 + threadIdx.x;   // B*XC*H*W threads
    const int w = idx & (Ww - 1);
    const int h = (idx >> 8) & (Hh - 1);
    const int c = (idx >> 16) & (XC - 1);
    const int b = idx >> 22;

    const long xoff  = ((long)(b * XC + c)) * Hh * Ww;
    const long kbase = (((long)(b * XC + c) * 9) * Hh + h) * Ww + w;

    float kv[9];
#pragma unroll
    for (int j = 0; j < 9; ++j) kv[j] = ker[kbase + (long)j * Hh * Ww];

    const float xc = x[xoff + h * Ww + w];
    float outs[3];
    const int dils[3] = {1, 3, 5};
#pragma unroll
    for (int br = 0; br < 3; ++br) {
        const int dil = dils[br];
        float acc = 0.f;
#pragma unroll
        for (int j = 0; j < 9; ++j) {
            const int hh = h + (j / 3 - 1) * dil;
            const int ww = w + (j % 3 - 1) * dil;
            float xv = 0.f;
            if ((unsigned)hh < (unsigned)Hh && (unsigned)ww < (unsigned)Ww)
                xv = x[xoff + hh * Ww + ww];
            acc += kv[j] * xv;
        }
        outs[br] = acc;
    }
    const long m0 = ((long)(b * MC + c) * Hh + h) * Ww + w;
    const long cs = (long)XC * Hh * Ww;
    merged[m0]          = xc;
    merged[m0 + cs]     = outs[0];
    merged[m0 + 2 * cs] = outs[1];
    merged[m0 + 3 * cs] = outs[2];
}

// ---------------------------------------------------------------------------
// Kernel 3a: repack fuse_w OIHW [64][256][3][3] -> [tap][ic][oc] so WMMA
// A-tiles stage contiguously.
// ---------------------------------------------------------------------------
__global__ __launch_bounds__(256) void repack_w_kernel(
    const float* __restrict__ fw, float* __restrict__ wt)
{
    const int idx = blockIdx.x * 256 + threadIdx.x;
    if (idx >= XC * MC * 9) return;
    const int oc  = idx / (MC * 9);
    const int rem = idx % (MC * 9);
    const int ic  = rem / 9;
    const int tap = rem % 9;
    wt[(tap * MC + ic) * XC + oc] = fw[idx];
}

// ---------------------------------------------------------------------------
// Kernel 3b: fuse 3x3 conv as implicit GEMM with V_WMMA_F32_16X16X4_F32.
// Block = (b, h, 32-wide w tile); 8 waves -> 4 oc-tiles x 2 pixel-subtiles.
// K = 256 ic x 9 taps, 4 at a time (576 WMMAs / tile). Both operands come
// from LDS: activations staged per ic-chunk, weight tile (64ic x 64oc, 16KB)
// staged per tap with coalesced float4 loads. ReLU + store.
// A (16x4 f32): lanes 0-15 hold K0/K1 in v[0]/v[1]; lanes 16-31 hold K2/K3.
// C/D (16x16 f32): v8f, VGPR r -> M=r (lanes 0-15) / M=r+8 (lanes 16-31).
// ---------------------------------------------------------------------------
__global__ __launch_bounds__(256) void fuse_conv_wmma_kernel(
    const float* __restrict__ merged, const float* __restrict__ wt,
    float* __restrict__ out)
{
    __shared__ float lds_act[64 * 3 * 34];   // 26112 B: [ic 64][row 3][col 34]
    __shared__ float lds_w[64 * 64];         // 16384 B: [ic 64][oc 64]

    const int blk   = blockIdx.x;
    const int wtile = blk & 7;                // 8 tiles of 32 pixels
    const int h     = (blk >> 3) & (Hh - 1);
    const int b     = blk >> 10;
    const int w0    = wtile * 32;

    const int tid   = threadIdx.x;
    const int wave  = tid >> 5;
    const int lane  = tid & 31;
    const int mt    = wave & 3;               // oc tile (0..3)
    const int nt    = wave >> 2;              // pixel sub-tile (0..1)
    const int lhalf = lane >> 4;              // 0: K0/K1, 1: K2/K3
    const int l16   = lane & 15;

    v8f acc = {};

    for (int icc = 0; icc < 4; ++icc) {
        const int ic0 = icc * 64;
        __syncthreads();                      // prior compute done before restage
        // --- stage activation chunk ---
        for (int i = tid; i < 64 * 3 * 34; i += 256) {
            const int col = i % 34;           // -> gw = w0-1+col
            const int r   = (i / 34) % 3;     // -> gh = h-1+r
            const int ch  = i / (34 * 3);
            const int gw  = w0 - 1 + col;
            const int gh  = h - 1 + r;
            float v = 0.f;
            if ((unsigned)gw < (unsigned)Ww && (unsigned)gh < (unsigned)Hh)
                v = merged[((long)(b * MC + ic0 + ch) * Hh + gh) * Ww + gw];
            lds_act[i] = v;
        }

        for (int kh = 0; kh < 3; ++kh) {
            for (int kw = 0; kw < 3; ++kw) {
                const int tap = kh * 3 + kw;
                __syncthreads();              // prior tap compute done; acts visible
                // --- stage weight tile [64 ic][64 oc] via float4 ---
                {
                    const float4* wsrc =
                        (const float4*)(wt + ((long)(tap * MC + ic0)) * XC);
                    float4* wdst = (float4*)lds_w;
                    for (int i = tid; i < 1024; i += 256)
                        wdst[i] = wsrc[i];
                }
                __syncthreads();

                const int col = nt * 16 + l16 + kw;        // 0..33
                const int aoc = mt * 16 + l16;
                for (int ks = 0; ks < 64; ks += 4) {
                    const int ka = ks + lhalf * 2;
                    v2f a, bf;
                    a.x = lds_w[(ka + 0) * 64 + aoc];
                    a.y = lds_w[(ka + 1) * 64 + aoc];
                    const int base = ka * (3 * 34) + kh * 34 + col;
                    bf.x = lds_act[base];
                    bf.y = lds_act[base + 3 * 34];         // next channel
                    acc = __builtin_amdgcn_wmma_f32_16x16x4_f32(
                        /*neg_a=*/false, a, /*neg_b=*/false, bf,
                        /*c_mod=*/(short)0, acc,
                        /*reuse_a=*/false, /*reuse_b=*/false);
                }
            }
        }
    }

    const int wpix = w0 + nt * 16 + l16;
#pragma unroll
    for (int r = 0; r < 8; ++r) {
        const int oc = mt * 16 + r + lhalf * 8;
        float v = acc[r];
        v = v > 0.f ? v : 0.f;                             // ReLU
        out[((long)(b * XC + oc) * Hh + h) * Ww + wpix] = v;
    }
}

// ---------------------------------------------------------------------------
extern "C" void kernel_launch(void* const* d_in, const int* in_sizes, int n_in,
                              void* d_out, int out_size, void* d_ws, size_t ws_size,
                              hipStream_t stream) {
    const float* ref = (const float*)d_in[0];
    const float* tgt = (const float*)d_in[1];
    const float* x   = (const float*)d_in[2];
    const float* ker = (const float*)d_in[3];
    const float* fw  = (const float*)d_in[4];
    // d_in[5] = maxdisp(48), d_in[6] = num_groups(40): compile-time constants.

    float* vol  = (float*)d_out;                                   // [2,40,48,128,256]
    float* outp = vol + (size_t)Bn * NG * MAXD * Hh * Ww;          // [2,64,128,256]

    float* merged = (float*)d_ws;                                  // [2,256,128,256]
    float* wrep   = merged + (size_t)Bn * MC * Hh * Ww;            // [9,256,64]

    // 1) cost volume: 2*128*4 blocks
    gwc_volume_kernel<<<Bn * Hh * 4, 256, 0, stream>>>(ref, tgt, vol);
    // 2) dynamic conv -> merged: 2*64*128*256 / 256 blocks
    dynconv_kernel<<<(Bn * XC * Hh * Ww) / 256, 256, 0, stream>>>(x, ker, merged);
    // 3a) weight repack
    repack_w_kernel<<<(XC * MC * 9 + 255) / 256, 256, 0, stream>>>(fw, wrep);
    // 3b) fused conv + ReLU via f32 WMMA: 2*128*8 blocks
    fuse_conv_wmma_kernel<<<Bn * Hh * 8, 256, 0, stream>>>(merged, wrep, outp);
}